// RD_loss9_58213986730403
// MI455X (gfx1250) — compile-verified
//
#include <hip/hip_runtime.h>
#include <hip/hip_bf16.h>

// Problem constants (from the reference):
//   pred, gt : [4, 1, 512, 512] float32
//   h_offsets, w_offsets : [90] int32, magnitudes in [1, 20]
//   SPAN = 20, REPEAT = 90, Hb = Wb = 472
#define SPAN   20
#define NOFF   90
#define IMG    512
#define HB     472          // IMG - 2*SPAN
#define TILE   32           // output tile edge per workgroup
#define REGN   72           // TILE + 2*SPAN  (staged region edge)
#define NTIL   15           // ceil(472 / 32)
#define NBATCH 4
#define NPART  (NTIL * NTIL * NBATCH)   // 900 partial sums

// gfx1250 async global->LDS builtins: b32 wants int* (AS1 src, AS3 dst).
typedef int   __attribute__((address_space(1))) gint;   // global int
typedef int   __attribute__((address_space(3))) lint;   // LDS int
typedef float v2f __attribute__((ext_vector_type(2)));  // 8B-aligned pair

__global__ __launch_bounds__(256) void rd_tile_kernel(
    const float* __restrict__ pred, const float* __restrict__ gt,
    const int* __restrict__ hoff, const int* __restrict__ woff,
    float* __restrict__ partial)
{
    // Interleaved tile: sPG[2*idx] = pred, sPG[2*idx+1] = gt  (40.5 KB).
    // One pixel's (pred, gt) pair = one ds_load_b64 -> all 64 LDS banks busy
    // per wave32 access, half the ds instruction count of split arrays.
    __shared__ float sPG[2 * REGN * REGN];
    __shared__ int   sD[NOFF];          // pre-doubled deltas 2*(h*72 + w)
    __shared__ float sRed[256];

    const int tid = threadIdx.x;
    const int i0  = blockIdx.y * TILE;          // tile origin, full-image coords
    const int j0  = blockIdx.x * TILE;
    const size_t imgbase = (size_t)blockIdx.z * (IMG * IMG);

    // Combined, pre-doubled shift offsets (interleaved-element units).
    if (tid < NOFF) sD[tid] = 2 * (hoff[tid] * REGN + woff[tid]);

    // ---- Async stage (GLOBAL_LOAD_ASYNC_TO_LDS_B32, tracked by ASYNCcnt):
    // per-element copies so pred/gt land interleaved. Consecutive lanes hit
    // consecutive global dwords -> fully coalesced. OOB rows/cols (edge tiles
    // only) clamp to duplicates that are read only by masked-off pixels.
    // Stride 256 = 3*72 + 40: update (row, col) incrementally, no div/mod in
    // the loop body.
    {
        int row = tid / REGN;               // one division at entry
        int col = tid - row * REGN;
        int v   = tid;                      // == row*REGN + col
        while (row < REGN) {
            int grow = i0 + row; grow = (grow > IMG - 1) ? (IMG - 1) : grow;
            int gcol = j0 + col; gcol = (gcol > IMG - 1) ? (IMG - 1) : gcol;
            const size_t goff = imgbase + (size_t)grow * IMG + gcol;
            __builtin_amdgcn_global_load_async_to_lds_b32(
                (gint*)(pred + goff), (lint*)&sPG[2 * v],     0, 0);
            __builtin_amdgcn_global_load_async_to_lds_b32(
                (gint*)(gt + goff),   (lint*)&sPG[2 * v + 1], 0, 0);
            v += 256; row += 3; col += 40;
            if (col >= REGN) { col -= REGN; row += 1; }
        }
    }
    __builtin_amdgcn_s_wait_asynccnt(0);   // this wave's async copies landed
    __syncthreads();                       // all waves' copies visible

    // ---- Compute: each thread owns 4 pixels, row-strided so a wave32 reads
    // ---- 32 consecutive float2s (256 B = 64 banks) -> conflict-free b64.
    const int tx = tid & 31;               // 0..31  -> local col
    const int ty = tid >> 5;               // 0..7   -> local row group

    int   ib[4];
    float pb[4], gb[4], sp[4], sg[4];
#pragma unroll
    for (int k = 0; k < 4; ++k) {
        const int il = ty + 8 * k;
        ib[k] = 2 * ((il + SPAN) * REGN + (tx + SPAN));
        const v2f b = *(const v2f*)&sPG[ib[k]];
        pb[k] = b.x; gb[k] = b.y;
        sp[k] = 0.0f; sg[k] = 0.0f;
    }

    // Pass 1: sum of squared diffs over the 90 shifts.
    for (int r = 0; r < NOFF; ++r) {
        const int d = sD[r];               // uniform -> LDS broadcast
#pragma unroll
        for (int k = 0; k < 4; ++k) {
            const v2f s = *(const v2f*)&sPG[ib[k] + d];
            const float dp = pb[k] - s.x;
            const float dg = gb[k] - s.y;
            sp[k] += dp * dp;
            sg[k] += dg * dg;
        }
    }

    float ip[4], ig[4];
#pragma unroll
    for (int k = 0; k < 4; ++k) {
        ip[k] = (sp[k] > 0.0f) ? (1.0f / sqrtf(sp[k])) : 1.0f;  // norm==0 -> 1
        ig[k] = (sg[k] > 0.0f) ? (1.0f / sqrtf(sg[k])) : 1.0f;
    }

    // Pass 2: recompute diffs from LDS, accumulate |dp/||p|| - dg/||g|||.
    float acc[4] = {0.0f, 0.0f, 0.0f, 0.0f};
    for (int r = 0; r < NOFF; ++r) {
        const int d = sD[r];
#pragma unroll
        for (int k = 0; k < 4; ++k) {
            const v2f s = *(const v2f*)&sPG[ib[k] + d];
            const float dp = pb[k] - s.x;
            const float dg = gb[k] - s.y;
            acc[k] += fabsf(dp * ip[k] - dg * ig[k]);
        }
    }

    // Validity mask for ragged last tiles (472 = 14*32 + 24).
    float total = 0.0f;
#pragma unroll
    for (int k = 0; k < 4; ++k) {
        const int gi = i0 + ty + 8 * k;
        const int gj = j0 + tx;
        if (gi < HB && gj < HB) total += acc[k];
    }

    // Deterministic block reduction -> one partial per workgroup.
    sRed[tid] = total;
    __syncthreads();
    for (int s = 128; s > 0; s >>= 1) {
        if (tid < s) sRed[tid] += sRed[tid + s];
        __syncthreads();
    }
    if (tid == 0)
        partial[((int)blockIdx.z * NTIL + (int)blockIdx.y) * NTIL + (int)blockIdx.x] = sRed[0];
}

// Final deterministic reduction of the 900 partials -> mean.
__global__ __launch_bounds__(256) void rd_final_reduce(
    const float* __restrict__ partial, float* __restrict__ out,
    int n, float scale)
{
    __shared__ float s[256];
    float a = 0.0f;
    for (int i = threadIdx.x; i < n; i += 256) a += partial[i];
    s[threadIdx.x] = a;
    __syncthreads();
    for (int st = 128; st > 0; st >>= 1) {
        if (threadIdx.x < st) s[threadIdx.x] += s[threadIdx.x + st];
        __syncthreads();
    }
    if (threadIdx.x == 0) out[0] = s[0] * scale;
}

extern "C" void kernel_launch(void* const* d_in, const int* in_sizes, int n_in,
                              void* d_out, int out_size, void* d_ws, size_t ws_size,
                              hipStream_t stream) {
    const float* pred = (const float*)d_in[0];
    const float* gt   = (const float*)d_in[1];
    const int*   ho   = (const int*)d_in[2];
    const int*   wo   = (const int*)d_in[3];
    float* out = (float*)d_out;
    float* ws  = (float*)d_ws;   // NPART floats of scratch (3.6 KB)

    dim3 grid(NTIL, NTIL, NBATCH);
    rd_tile_kernel<<<grid, 256, 0, stream>>>(pred, gt, ho, wo, ws);

    // mean over B*Hb*Wb*R = 4*472*472*90 = 80,202,240 elements
    const float scale = 1.0f / 80202240.0f;
    rd_final_reduce<<<1, 256, 0, stream>>>(ws, out, NPART, scale);
}